// SimpleGraphSAGE_63367947485322
// MI455X (gfx1250) — compile-verified
//
#include <hip/hip_runtime.h>

typedef float v2f __attribute__((ext_vector_type(2)));
typedef float v8f __attribute__((ext_vector_type(8)));

#define N_NODES 50000
#define N_EDGES 640000
#define IN_CH   128
#define HID_CH  256

// ---------------- Phase 0: zero the aggregation workspace ----------------
__global__ void sage_zero(float* __restrict__ ws, int n) {
  int i = blockIdx.x * blockDim.x + threadIdx.x;
  int stride = gridDim.x * blockDim.x;
  for (; i < n; i += stride) ws[i] = 0.0f;
}

// ---------------- Phase 1: edge scatter (mean-aggregation numerator + counts)
// One wave32 per edge: lane L moves channels [4L, 4L+4) as a float4.
// x is L2-resident (25.6MB << 192MB L2), so the random gather hits L2.
__global__ void sage_scatter(const float* __restrict__ x,
                             const long long* __restrict__ ei,   // [2, E] int64
                             float* __restrict__ agg,
                             float* __restrict__ cnt) {
  int gid  = blockIdx.x * blockDim.x + threadIdx.x;
  int edge = gid >> 5;
  if (edge >= N_EDGES) return;
  int lane = gid & 31;
  long long s = ei[edge];
  long long d = ei[N_EDGES + edge];
  const float4 v = *reinterpret_cast<const float4*>(x + s * IN_CH + lane * 4);
  float* a = agg + d * IN_CH + lane * 4;
  atomicAdd(a + 0, v.x);
  atomicAdd(a + 1, v.y);
  atomicAdd(a + 2, v.z);
  atomicAdd(a + 3, v.w);
  if (lane == 0) atomicAdd(cnt + d, 1.0f);
}

// ---------------- Phase 2: fused GEMM  out = mean @ W_l + x @ W_r + b_l
// One wave computes a 16x64 output slab via V_WMMA_F32_16X16X4_F32,
// K = 256 (first 128 from agg*inv_cnt against W_l, next 128 from x against W_r).
__global__ __launch_bounds__(256)
void sage_gemm(const float* __restrict__ x,
               const float* __restrict__ agg,
               const float* __restrict__ cnt,
               const float* __restrict__ Wl,   // [128,256] row-major
               const float* __restrict__ bl,   // [256]
               const float* __restrict__ Wr,   // [128,256] row-major
               float* __restrict__ out) {      // [50000,256]
  constexpr int NT = 4;                               // 4 n-tiles of 16 => 64 cols/wave
  constexpr int NBLK = HID_CH / (16 * NT);            // 4 column-slabs
  constexpr int WAVES = (N_NODES / 16) * NBLK;        // 3125 * 4 = 12500

  int wave = (blockIdx.x * blockDim.x + threadIdx.x) >> 5;
  if (wave >= WAVES) return;                          // wave-uniform: EXEC stays all-1s

  int lane = threadIdx.x & 31;
  int half = lane >> 4;                               // selects K pair within the 4-step
  int l16  = lane & 15;

  int m0 = (wave / NBLK) * 16;
  int n0 = (wave % NBLK) * (16 * NT);

  // Per-lane A-row and its 1/max(count,1) (A-matrix: lanes 0-15 and 16-31 both hold M=l16)
  int rowA = m0 + l16;
  float inv = 1.0f / fmaxf(cnt[rowA], 1.0f);

  v8f acc[NT] = {};

  // ---- K = 0..127 : (agg * inv) @ W_l
  const float* arow = agg + (long)rowA * IN_CH;
  #pragma unroll 4
  for (int k = 0; k < IN_CH; k += 4) {
    int kk = k + 2 * half;                 // lanes 0-15: K={k,k+1}; lanes 16-31: K={k+2,k+3}
    v2f a;
    a.x = arow[kk]     * inv;
    a.y = arow[kk + 1] * inv;
    const float* w0 = Wl + (long)kk * HID_CH;
    const float* w1 = w0 + HID_CH;
    #pragma unroll
    for (int j = 0; j < NT; ++j) {
      int col = n0 + 16 * j + l16;
      v2f b;
      b.x = w0[col];
      b.y = w1[col];
      acc[j] = __builtin_amdgcn_wmma_f32_16x16x4_f32(
          /*neg_a=*/false, a, /*neg_b=*/false, b,
          /*c_mod=*/(short)0, acc[j], /*reuse_a=*/false, /*reuse_b=*/false);
    }
  }

  // ---- K = 128..255 : x @ W_r
  const float* xrow = x + (long)rowA * IN_CH;
  #pragma unroll 4
  for (int k = 0; k < IN_CH; k += 4) {
    int kk = k + 2 * half;
    v2f a;
    a.x = xrow[kk];
    a.y = xrow[kk + 1];
    const float* w0 = Wr + (long)kk * HID_CH;
    const float* w1 = w0 + HID_CH;
    #pragma unroll
    for (int j = 0; j < NT; ++j) {
      int col = n0 + 16 * j + l16;
      v2f b;
      b.x = w0[col];
      b.y = w1[col];
      acc[j] = __builtin_amdgcn_wmma_f32_16x16x4_f32(
          false, a, false, b, (short)0, acc[j], false, false);
    }
  }

  // ---- Store with bias. C/D layout: VGPR r -> row m0 + r + 8*half, col n0 + 16j + l16
  #pragma unroll
  for (int j = 0; j < NT; ++j) {
    int col = n0 + 16 * j + l16;
    float bias = bl[col];
    #pragma unroll
    for (int r = 0; r < 8; ++r) {
      int row = m0 + r + 8 * half;
      out[(long)row * HID_CH + col] = acc[j][r] + bias;
    }
  }
}

extern "C" void kernel_launch(void* const* d_in, const int* in_sizes, int n_in,
                              void* d_out, int out_size, void* d_ws, size_t ws_size,
                              hipStream_t stream) {
  const float*     x  = (const float*)d_in[0];       // [50000,128] fp32
  const long long* ei = (const long long*)d_in[1];   // [2,640000] int64
  const float*     Wl = (const float*)d_in[2];       // [128,256]
  const float*     bl = (const float*)d_in[3];       // [256]
  const float*     Wr = (const float*)d_in[4];       // [128,256]
  float* out = (float*)d_out;                        // [50000,256]

  float* agg = (float*)d_ws;                         // [50000*128]
  float* cnt = agg + (size_t)N_NODES * IN_CH;        // [50000]

  // Phase 0: zero agg+cnt (contiguous in ws)
  int zn = N_NODES * IN_CH + N_NODES;
  sage_zero<<<(zn + 255) / 256, 256, 0, stream>>>(agg, zn);

  // Phase 1: edge scatter, one wave32 per edge
  long long threads = (long long)N_EDGES * 32;
  sage_scatter<<<(int)((threads + 255) / 256), 256, 0, stream>>>(x, ei, agg, cnt);

  // Phase 2: WMMA GEMM, 12500 waves (8 waves per 256-thread block)
  int waves  = (N_NODES / 16) * (HID_CH / 64);       // 12500
  int blocks = (waves * 32 + 255) / 256;             // 1563
  sage_gemm<<<blocks, 256, 0, stream>>>(x, agg, cnt, Wl, bl, Wr, out);
}